// GCN_14654428414705
// MI455X (gfx1250) — compile-verified
//
#include <hip/hip_runtime.h>
#include <hip/hip_bf16.h>

typedef __attribute__((ext_vector_type(2))) float v2f;
typedef __attribute__((ext_vector_type(8))) float v8f;

#define IN_FT   128
#define OUT_FT  128
#define SELF_W  3.0f

// ---------------------------------------------------------------------------
// Kernel 1: x = seq @ W^T via V_WMMA_F32_16X16X4_F32.
// One wave computes a 16-row x 128-col strip (8 accumulators of 16x16).
// A frag (16x4):  lane<16 -> (M=lane,  K=kb+0..1), lane>=16 -> (M=lane-16, K=kb+2..3)
// B frag (4x16):  lane<16 -> (K=kb+0..1, N=nb+lane), lane>=16 -> (K=kb+2..3, N=nb+lane-16)
// B[k][n] = W[n][k]  (W row-major [OUT][IN]), consecutive k -> contiguous float2.
// C/D (16x16 f32): VGPR i -> M = mbase + i + (lane>=16 ? 8 : 0), N = nb + (lane&15)
// ---------------------------------------------------------------------------
__global__ __launch_bounds__(256) void gcn_gemm_wmma(
    const float* __restrict__ seq, const float* __restrict__ W,
    float* __restrict__ x, int n_rows) {
  const int lane  = threadIdx.x & 31;
  const int wave  = threadIdx.x >> 5;
  const int waves_per_block = blockDim.x >> 5;
  const int mbase = (blockIdx.x * waves_per_block + wave) * 16;
  if (mbase >= n_rows) return;   // wave-uniform: EXEC stays all-ones for WMMA

  const int half = lane >> 4;    // 0: K pair {0,1}; 1: K pair {2,3}
  const int l16  = lane & 15;

  v8f acc[8];
#pragma unroll
  for (int t = 0; t < 8; ++t) acc[t] = (v8f){};

  const float* seq_row = seq + (size_t)(mbase + l16) * IN_FT;

#pragma unroll 4
  for (int kb = 0; kb < IN_FT; kb += 4) {
    const int k0 = kb + half * 2;
    const v2f a = *(const v2f*)(seq_row + k0);           // 8B aligned (k0 even)
#pragma unroll
    for (int t = 0; t < 8; ++t) {
      const int ncol = t * 16 + l16;
      const v2f b = *(const v2f*)(W + (size_t)ncol * IN_FT + k0);
      acc[t] = __builtin_amdgcn_wmma_f32_16x16x4_f32(
          /*neg_a=*/false, a, /*neg_b=*/false, b,
          /*c_mod=*/(short)0, acc[t], /*reuse_a=*/false, /*reuse_b=*/false);
    }
  }

#pragma unroll
  for (int t = 0; t < 8; ++t) {
    const int col = t * 16 + l16;
#pragma unroll
    for (int i = 0; i < 8; ++i) {
      const int row = mbase + i + half * 8;
      x[(size_t)row * OUT_FT + col] = acc[t][i];
    }
  }
}

// ---------------------------------------------------------------------------
// Degree / normalization setup
// ---------------------------------------------------------------------------
__global__ void gcn_deg_init(float* __restrict__ deg, int n) {
  int i = blockIdx.x * blockDim.x + threadIdx.x;
  if (i < n) deg[i] = SELF_W;                     // self-loop fill contribution
}

__global__ void gcn_deg_accum(const int* __restrict__ edge_index,
                              const float* __restrict__ edge_weight,
                              float* __restrict__ deg, int num_e) {
  int e = blockIdx.x * blockDim.x + threadIdx.x;
  if (e < num_e) atomicAdd(&deg[edge_index[e]], edge_weight[e]);  // row = edge_index[0]
}

__global__ void gcn_deg_finalize(float* __restrict__ deg, int n) {
  int i = blockIdx.x * blockDim.x + threadIdx.x;
  if (i < n) {
    float d = deg[i];
    deg[i] = (d > 0.0f) ? rsqrtf(d) : 0.0f;       // in-place -> dinv
  }
}

// ---------------------------------------------------------------------------
// out[n][f] = 3 * dinv[n]^2 * x[n][f]   (self-loop term; initializes d_out)
// ---------------------------------------------------------------------------
__global__ void gcn_out_init(const float* __restrict__ x,
                             const float* __restrict__ dinv,
                             float* __restrict__ out, int total) {
  int i = blockIdx.x * blockDim.x + threadIdx.x;
  if (i < total) {
    float d = dinv[i >> 7];                        // OUT_FT == 128
    out[i] = SELF_W * d * d * x[i];
  }
}

// ---------------------------------------------------------------------------
// One wave32 per edge: lane moves float4 -> 32*16B = 512B/row, fully coalesced.
// out[r][:] += dinv[r]*w*dinv[c] * x[c][:]  via global_atomic_add_f32.
// ---------------------------------------------------------------------------
__global__ __launch_bounds__(256) void gcn_edge_agg(
    const int* __restrict__ edge_index, const float* __restrict__ edge_weight,
    const float* __restrict__ x, const float* __restrict__ dinv,
    float* __restrict__ out, int num_e) {
  const int e    = (blockIdx.x * blockDim.x + threadIdx.x) >> 5;
  const int lane = threadIdx.x & 31;
  if (e >= num_e) return;

  const int r = edge_index[e];
  const int c = edge_index[num_e + e];
  const float nrm = dinv[r] * edge_weight[e] * dinv[c];

  const float4 v = ((const float4*)(x + (size_t)c * OUT_FT))[lane];
  float* o = out + (size_t)r * OUT_FT + lane * 4;
  atomicAdd(o + 0, nrm * v.x);
  atomicAdd(o + 1, nrm * v.y);
  atomicAdd(o + 2, nrm * v.z);
  atomicAdd(o + 3, nrm * v.w);
}

__global__ void gcn_relu(float* __restrict__ out, int total) {
  int i = blockIdx.x * blockDim.x + threadIdx.x;
  if (i < total) out[i] = fmaxf(out[i], 0.0f);
}

// ---------------------------------------------------------------------------
// Host-side launch
// Inputs (setup_inputs order): 0=seq [N*128 f32], 1=edge_weight [E f32],
//                              2=W [128*128 f32], 3=edge_index [2*E int32]
// ---------------------------------------------------------------------------
extern "C" void kernel_launch(void* const* d_in, const int* in_sizes, int n_in,
                              void* d_out, int out_size, void* d_ws, size_t ws_size,
                              hipStream_t stream) {
  const float* seq         = (const float*)d_in[0];
  const float* edge_weight = (const float*)d_in[1];
  const float* W           = (const float*)d_in[2];
  const int*   edge_index  = (const int*)d_in[3];

  const int N = in_sizes[0] / IN_FT;
  const int E = in_sizes[1];
  const int total = N * OUT_FT;

  float* out = (float*)d_out;

  // Workspace layout: x [N*OUT_FT f32], then deg/dinv [N f32]
  float* x   = (float*)d_ws;
  float* deg = x + (size_t)N * OUT_FT;

  // 1) x = seq @ W^T  (fp32 WMMA; 16 rows per wave, 8 waves per block)
  {
    const int rows_per_block = 8 * 16;
    const int blocks = (N + rows_per_block - 1) / rows_per_block;
    gcn_gemm_wmma<<<blocks, 256, 0, stream>>>(seq, W, x, N);
  }

  // 2) degree -> dinv
  gcn_deg_init<<<(N + 255) / 256, 256, 0, stream>>>(deg, N);
  gcn_deg_accum<<<(E + 255) / 256, 256, 0, stream>>>(edge_index, edge_weight, deg, E);
  gcn_deg_finalize<<<(N + 255) / 256, 256, 0, stream>>>(deg, N);

  // 3) self-loop term initializes out
  gcn_out_init<<<(total + 255) / 256, 256, 0, stream>>>(x, deg, out, total);

  // 4) edge scatter-add (one wave per edge)
  {
    const long long threads = (long long)E * 32;
    const int blocks = (int)((threads + 255) / 256);
    gcn_edge_agg<<<blocks, 256, 0, stream>>>(edge_index, edge_weight, x, deg, out, E);
  }

  // 5) ReLU
  gcn_relu<<<(total + 255) / 256, 256, 0, stream>>>(out, total);
}